// AdaRegSpecLoss_16647293240106
// MI455X (gfx1250) — compile-verified
//
#include <hip/hip_runtime.h>

// AdaRegSpecLoss (blockwise Tversky) for MI455X / gfx1250.
// Memory-bound: 262 MB read @ 23.3 TB/s => ~11 us floor. Strategy:
//   kernel 1: one 256-thread workgroup per 16^3 block; float4 streaming loads,
//             per-wave reduction on the matrix pipe via V_WMMA_F32_16X16X4_F32
//             (A = data, B = ones => D accumulates row sums; 64 vals/WMMA),
//             per-block loss written to workspace (no global atomics).
//   kernel 2: single workgroup reduces the 8000 per-block losses into d_out[0].

#define TV_PZ     16
#define TV_NBLK   8000          // 4 * 2 * 10 * 10 * 10
#define TV_ZSTR   (160 * 160)
#define TV_XSTR   160
#define TV_SMOOTH 1e-8f

typedef __attribute__((ext_vector_type(2))) float v2f;
typedef __attribute__((ext_vector_type(8))) float v8f;

#if defined(__gfx1250__) && __has_builtin(__builtin_amdgcn_wmma_f32_16x16x4_f32)
#define TV_USE_WMMA 1
#else
#define TV_USE_WMMA 0
#endif

#if TV_USE_WMMA
// D = A(16x4) * ones(4x16) + C : every D[m,n] accumulates a row-sum of A,
// so one call folds all 64 fp32 values held by the wave into C.
__device__ __forceinline__ v8f tv_wmma_rowsum(v2f a, v2f ones, v8f c) {
  return __builtin_amdgcn_wmma_f32_16x16x4_f32(
      /*neg_a=*/false, a, /*neg_b=*/false, ones,
      /*c_mod=*/(short)0, c, /*reuse_a=*/false, /*reuse_b=*/false);
}
#endif

__global__ __launch_bounds__(256) void tversky_block_kernel(
    const float* __restrict__ pred, const float* __restrict__ gt,
    const float* __restrict__ pa, const float* __restrict__ pb,
    float* __restrict__ block_out, int atomic_mode) {
  const int b   = blockIdx.x;
  const int tid = threadIdx.x;

  // block b -> (n, c, zz, xx, yy); summation order is irrelevant to the result
  const int yy = b % 10;
  const int xx = (b / 10) % 10;
  const int zz = (b / 100) % 10;
  const int c  = (b / 1000) % 2;
  const int n  = b / 2000;
  const size_t base =
      (((size_t)(n * 2 + c) * 160 + (size_t)zz * TV_PZ) * 160 + (size_t)xx * TV_PZ) * 160 +
      (size_t)yy * TV_PZ;

#if TV_USE_WMMA
  v8f ct = {}, cg = {}, cp = {};
  v2f ones; ones[0] = 1.0f; ones[1] = 1.0f;
#else
  float st = 0.0f, sg = 0.0f, sp = 0.0f;
#endif

#pragma unroll
  for (int k = 0; k < 4; ++k) {
    const int idx4 = tid + (k << 8);          // [0, 1024) float4 index in block
    const int y4 = idx4 & 3;
    const int x  = (idx4 >> 2) & 15;
    const int z  = idx4 >> 6;
    const size_t off = base + (size_t)z * TV_ZSTR + (size_t)x * TV_XSTR + (size_t)y4 * 4;

    const float4 p4 = *reinterpret_cast<const float4*>(pred + off);  // global_load_b128
    const float4 g4 = *reinterpret_cast<const float4*>(gt + off);    // global_load_b128

    const float t0 = g4.x * p4.x, t1 = g4.y * p4.y;
    const float t2 = g4.z * p4.z, t3 = g4.w * p4.w;

#if TV_USE_WMMA
    v2f a;
    a[0] = t0;   a[1] = t1;   ct = tv_wmma_rowsum(a, ones, ct);
    a[0] = t2;   a[1] = t3;   ct = tv_wmma_rowsum(a, ones, ct);
    a[0] = g4.x; a[1] = g4.y; cg = tv_wmma_rowsum(a, ones, cg);
    a[0] = g4.z; a[1] = g4.w; cg = tv_wmma_rowsum(a, ones, cg);
    a[0] = p4.x; a[1] = p4.y; cp = tv_wmma_rowsum(a, ones, cp);
    a[0] = p4.z; a[1] = p4.w; cp = tv_wmma_rowsum(a, ones, cp);
#else
    st += t0 + t1 + t2 + t3;
    sg += g4.x + g4.y + g4.z + g4.w;
    sp += p4.x + p4.y + p4.z + p4.w;
#endif
  }

  // ---- per-wave totals (wave32) ----
  float wt, wg, wp;
#if TV_USE_WMMA
  // lane L holds C[M = j + (L<16 ? 0 : 8), N = L%16] in component j; all N equal.
  wt = ct[0] + ct[1] + ct[2] + ct[3] + ct[4] + ct[5] + ct[6] + ct[7];
  wg = cg[0] + cg[1] + cg[2] + cg[3] + cg[4] + cg[5] + cg[6] + cg[7];
  wp = cp[0] + cp[1] + cp[2] + cp[3] + cp[4] + cp[5] + cp[6] + cp[7];
  wt += __shfl_xor(wt, 16, 32);  // lanes 0-15 hold rows 0-7, 16-31 hold rows 8-15
  wg += __shfl_xor(wg, 16, 32);
  wp += __shfl_xor(wp, 16, 32);
#else
  wt = st; wg = sg; wp = sp;
#pragma unroll
  for (int o = 16; o > 0; o >>= 1) {
    wt += __shfl_xor(wt, o, 32);
    wg += __shfl_xor(wg, o, 32);
    wp += __shfl_xor(wp, o, 32);
  }
#endif

  // ---- cross-wave (8 waves) + per-block loss ----
  __shared__ float ls[8][4];
  const int wave = tid >> 5;
  if ((tid & 31) == 0) { ls[wave][0] = wt; ls[wave][1] = wg; ls[wave][2] = wp; }
  __syncthreads();

  if (tid == 0) {
    float tp = 0.0f, gs = 0.0f, ps = 0.0f;
#pragma unroll
    for (int w = 0; w < 8; ++w) { tp += ls[w][0]; gs += ls[w][1]; ps += ls[w][2]; }
    const float fn = gs - tp;
    const float fp = ps - tp;
    const float denom = fp + fn + TV_SMOOTH;
    const float av = pa[0];
    const float bv = pb[0];
    const float alpha = av + bv * ((fp + TV_SMOOTH) / denom);
    const float beta  = av + bv * ((fn + TV_SMOOTH) / denom);
    const float contrib =
        1.0f - (tp + TV_SMOOTH) / (tp + alpha * fp + beta * fn + TV_SMOOTH);
    if (atomic_mode) {
      atomicAdd(&block_out[0], contrib);
    } else {
      block_out[b] = contrib;
    }
  }
}

__global__ __launch_bounds__(256) void tversky_reduce_kernel(
    const float* __restrict__ in, float* __restrict__ out, int n) {
  __shared__ float ls[256];
  float s = 0.0f;
  for (int i = threadIdx.x; i < n; i += 256) s += in[i];
  ls[threadIdx.x] = s;
  __syncthreads();
#pragma unroll
  for (int stride = 128; stride > 0; stride >>= 1) {
    if ((int)threadIdx.x < stride) ls[threadIdx.x] += ls[threadIdx.x + stride];
    __syncthreads();
  }
  if (threadIdx.x == 0) out[0] = ls[0];
}

__global__ void tversky_zero_kernel(float* __restrict__ out) {
  if (threadIdx.x == 0) out[0] = 0.0f;
}

extern "C" void kernel_launch(void* const* d_in, const int* in_sizes, int n_in,
                              void* d_out, int out_size, void* d_ws, size_t ws_size,
                              hipStream_t stream) {
  (void)in_sizes; (void)n_in; (void)out_size;
  const float* pred = (const float*)d_in[0];
  const float* gt   = (const float*)d_in[1];
  const float* pa   = (const float*)d_in[2];
  const float* pb   = (const float*)d_in[3];
  float* out = (float*)d_out;
  float* ws  = (float*)d_ws;

  if (ws_size >= (size_t)TV_NBLK * sizeof(float)) {
    tversky_block_kernel<<<TV_NBLK, 256, 0, stream>>>(pred, gt, pa, pb, ws, 0);
    tversky_reduce_kernel<<<1, 256, 0, stream>>>(ws, out, TV_NBLK);
  } else {
    // fallback: accumulate straight into d_out (zeroed first; d_out is poisoned)
    tversky_zero_kernel<<<1, 32, 0, stream>>>(out);
    tversky_block_kernel<<<TV_NBLK, 256, 0, stream>>>(pred, gt, pa, pb, out, 1);
  }
}